// FairnessLoss_63230508532295
// MI455X (gfx1250) — compile-verified
//
#include <hip/hip_runtime.h>
#include <hip/hip_bf16.h>

typedef __attribute__((ext_vector_type(2))) float v2f;
typedef __attribute__((ext_vector_type(8))) float v8f;

#define B_N 2048
#define D_N 512
#define G_N 8
#define NEGW (-1.0e30f)
#define LOGB 7.6246189861593985f /* log(2048) */

__device__ __forceinline__ void wait_asynccnt0() {
#if __has_builtin(__builtin_amdgcn_s_wait_asynccnt)
  __builtin_amdgcn_s_wait_asynccnt(0);
#else
  asm volatile("s_wait_asynccnt 0" ::: "memory");
#endif
}

// ---------------- squared norms: one block per row ----------------
__global__ __launch_bounds__(256) void sqnorm_kernel(const float* __restrict__ X,
                                                     float* __restrict__ sqn) {
  __shared__ float part[8];
  const int row = blockIdx.x, tid = threadIdx.x;
  float s = 0.0f;
  for (int u = tid; u < D_N; u += 256) { float v = X[row * D_N + u]; s += v * v; }
  for (int off = 16; off >= 1; off >>= 1) s += __shfl_xor(s, off, 32);
  if ((tid & 31) == 0) part[tid >> 5] = s;
  __syncthreads();
  if (tid == 0) { float t = 0.0f; for (int i = 0; i < 8; ++i) t += part[i]; sqn[row] = t; }
}

// ---------------- subgroup counts + init diam-max ----------------
__global__ __launch_bounds__(256) void counts_kernel(const int* __restrict__ sub,
                                                     int* __restrict__ counts,
                                                     float* __restrict__ logcnt,
                                                     unsigned* __restrict__ dmax) {
  __shared__ int c[G_N];
  const int tid = threadIdx.x;
  if (tid < G_N) c[tid] = 0;
  __syncthreads();
  for (int j = tid; j < B_N; j += 256) {
    int g = sub[j];
    if (g >= 0 && g < G_N) atomicAdd(&c[g], 1);
  }
  __syncthreads();
  if (tid < G_N) {
    int cg = c[tid];
    counts[tid] = cg;
    logcnt[tid] = __logf((float)(cg > 1 ? cg : 1));
  }
  if (tid == 0) *dmax = 0u;
}

// ---------------- WMMA fp32 GEMM: C = 0.5*max(n_i + n_j - 2 x.x^T, 0) ----------------
// One wave per 16x16 tile; K stepped by 4 with V_WMMA_F32_16X16X4_F32.
// A 16x4 layout: lanes 0-15 M=0..15 hold {K0,K1}; lanes 16-31 hold {K2,K3}.
__global__ __launch_bounds__(32) void gemm_cost_kernel(const float* __restrict__ X,
                                                       const float* __restrict__ sqn,
                                                       float* __restrict__ C,
                                                       unsigned* __restrict__ dmax) {
  const int lane = threadIdx.x;
  const int half = lane >> 4;      // 0 -> K pair {0,1}, 1 -> {2,3}
  const int l    = lane & 15;
  const int tm = blockIdx.x * 16;
  const int tn = blockIdx.y * 16;
  const float2* X2 = (const float2*)X;
  const int arow = (tm + l) * (D_N / 2);
  const int brow = (tn + l) * (D_N / 2);
  v8f acc = {};
  for (int kk2 = 0; kk2 < D_N / 2; kk2 += 2) {   // kk2 = K/2, step 2 -> K step 4
    float2 av = X2[arow + kk2 + half];
    float2 bv = X2[brow + kk2 + half];
    v2f a; a.x = av.x; a.y = av.y;
    v2f b; b.x = bv.x; b.y = bv.y;
    acc = __builtin_amdgcn_wmma_f32_16x16x4_f32(false, a, false, b,
                                                (short)0, acc, false, false);
  }
  float mx = 0.0f;
  #pragma unroll
  for (int v = 0; v < 8; ++v) {
    const int mrow = tm + v + 8 * half;  // C/D layout: VGPR v -> M=v (lanes<16) / v+8
    const int ncol = tn + l;
    float sq = sqn[mrow] + sqn[ncol] - 2.0f * acc[v];
    float cost = 0.5f * fmaxf(sq, 0.0f);
    C[mrow * B_N + ncol] = cost;
    mx = fmaxf(mx, cost);
  }
  for (int off = 16; off >= 1; off >>= 1) mx = fmaxf(mx, __shfl_xor(mx, off, 32));
  if (lane == 0) atomicMax(dmax, __float_as_uint(mx)); // cost >= 0 -> bit-monotonic
}

// ---------------- eps schedule on device ----------------
__global__ void eps_kernel(const unsigned* __restrict__ dmax, float* __restrict__ eps_sched) {
  const int t = threadIdx.x;
  if (t < 64) {
    float cmax = __uint_as_float(*dmax);
    float diam = sqrtf(2.0f * fmaxf(cmax, 1e-12f));
    float sigma = fmaxf(diam * powf(0.9f, (float)t), 0.05f);
    eps_sched[t] = sigma * sigma;
  }
}

// ---------------- zero init (g, fxx0, fyy0 are contiguous) ----------------
__global__ void zero_kernel(float* __restrict__ p, int n) {
  int i = blockIdx.x * blockDim.x + threadIdx.x;
  if (i < n) p[i] = 0.0f;
}

// ---------------- precompute weight vectors w[k][j] = logw + pot/eps ----------------
// mode 0 (phase A, K=17): k<8 -> lb + g/eps ; k<16 -> la + fxx/eps ; k==16 -> lb + fyy/eps
// mode 1 (phase B, K=8):  la + f/eps
__global__ __launch_bounds__(256) void prepw_kernel(const float* __restrict__ eps_sched, int t,
                                                    int mode,
                                                    const float* __restrict__ pot,      // g (A) or f (B)
                                                    const float* __restrict__ fxx_cur,
                                                    const float* __restrict__ fyy_cur,
                                                    const int* __restrict__ sub,
                                                    const float* __restrict__ logcnt,
                                                    float* __restrict__ w) {
  const int idx = blockIdx.x * blockDim.x + threadIdx.x;
  const int k = idx >> 11;          // /2048
  const int j = idx & (B_N - 1);
  const float inv_eps = 1.0f / eps_sched[t];
  float val;
  if (mode == 0) {
    if (k < 8) {
      val = -LOGB + pot[k * B_N + j] * inv_eps;
    } else if (k < 16) {
      const int gi = k - 8;
      float lw = (sub[j] == gi) ? -logcnt[gi] : NEGW;
      val = lw + fxx_cur[gi * B_N + j] * inv_eps;
    } else {
      val = -LOGB + fyy_cur[j] * inv_eps;
    }
  } else {
    float lw = (sub[j] == k) ? -logcnt[k] : NEGW;
    val = lw + pot[k * B_N + j] * inv_eps;
  }
  w[k * B_N + j] = val;
}

// ---------------- row-tiled LSE sweep over C ----------------
// Block handles ROWTILE=4 rows of C (staged to LDS via async global->LDS B128, 32KB)
// and all K weight slots.  out_i = -eps*(M + log(sum_j exp(w_j - C[i,j]/eps - M)))
__global__ __launch_bounds__(256) void lse_kernel(const float* __restrict__ C,
                                                  const float* __restrict__ w,
                                                  const float* __restrict__ eps_sched, int t,
                                                  int K, int mode,
                                                  float* __restrict__ f_or_g,
                                                  const float* __restrict__ fxx_cur,
                                                  float* __restrict__ fxx_next,
                                                  const float* __restrict__ fyy_cur,
                                                  float* __restrict__ fyy_next) {
  __shared__ float cbuf[4 * B_N];
  __shared__ float pm[8][4];
  __shared__ float ps[8][4];
  const int tid = threadIdx.x;
  const int r0 = blockIdx.x * 4;
  const float eps = eps_sched[t];
  const float inv_eps = 1.0f / eps;

  // stage 4 rows of C (32KB) into LDS with async global->LDS b128 (ASYNCcnt path):
  // low 32 bits of a generic pointer to __shared__ == wave-relative LDS byte offset.
  {
    const unsigned lds_base = (unsigned)(uintptr_t)&cbuf[0];
    const float* gbase = C + (size_t)r0 * B_N;
    #pragma unroll
    for (int u = 0; u < (4 * B_N / 4) / 256; ++u) {     // 8 b128 transfers per lane
      const int e = tid + u * 256;                      // float4 index
      const unsigned lds_addr = lds_base + (unsigned)e * 16u;
      const float* gptr = gbase + e * 4;
      asm volatile("global_load_async_to_lds_b128 %0, %1, off"
                   :: "v"(lds_addr), "v"(gptr) : "memory");
    }
    wait_asynccnt0();
  }
  __syncthreads();
  const int wave = tid >> 5, lane = tid & 31;

  for (int k = 0; k < K; ++k) {
    // cache my 8 w values (reused by both passes, shared across 4 rows)
    float wv[8];
    #pragma unroll
    for (int u = 0; u < 8; ++u) wv[u] = w[k * B_N + tid + 256 * u];

    // ---- pass 1: max ----
    float m[4];
    #pragma unroll
    for (int r = 0; r < 4; ++r) m[r] = -3.0e38f;
    #pragma unroll
    for (int u = 0; u < 8; ++u) {
      const int j = tid + 256 * u;
      #pragma unroll
      for (int r = 0; r < 4; ++r)
        m[r] = fmaxf(m[r], wv[u] - cbuf[r * B_N + j] * inv_eps);
    }
    #pragma unroll
    for (int r = 0; r < 4; ++r)
      for (int off = 16; off >= 1; off >>= 1)
        m[r] = fmaxf(m[r], __shfl_xor(m[r], off, 32));
    if (lane == 0) {
      #pragma unroll
      for (int r = 0; r < 4; ++r) pm[wave][r] = m[r];
    }
    __syncthreads();
    float M[4];
    #pragma unroll
    for (int r = 0; r < 4; ++r) {
      float mm = pm[0][r];
      #pragma unroll
      for (int q = 1; q < 8; ++q) mm = fmaxf(mm, pm[q][r]);
      M[r] = mm;
    }

    // ---- pass 2: sum of exp ----
    float s[4] = {0.0f, 0.0f, 0.0f, 0.0f};
    #pragma unroll
    for (int u = 0; u < 8; ++u) {
      const int j = tid + 256 * u;
      #pragma unroll
      for (int r = 0; r < 4; ++r)
        s[r] += __expf(wv[u] - cbuf[r * B_N + j] * inv_eps - M[r]);
    }
    #pragma unroll
    for (int r = 0; r < 4; ++r)
      for (int off = 16; off >= 1; off >>= 1)
        s[r] += __shfl_xor(s[r], off, 32);
    if (lane == 0) {
      #pragma unroll
      for (int r = 0; r < 4; ++r) ps[wave][r] = s[r];
    }
    __syncthreads();

    if (tid < 4) {
      const int r = tid;
      float S = 0.0f;
      #pragma unroll
      for (int q = 0; q < 8; ++q) S += ps[q][r];
      const int i = r0 + r;
      const float res = -eps * (M[r] + __logf(S));
      if (mode == 0) {
        if (k < 8)       f_or_g[k * B_N + i] = res;
        else if (k < 16) { int gi = k - 8;
                           fxx_next[gi * B_N + i] = 0.5f * (fxx_cur[gi * B_N + i] + res); }
        else             fyy_next[i] = 0.5f * (fyy_cur[i] + res);
      } else {
        f_or_g[k * B_N + i] = res;
      }
    }
  }
}

// ---------------- finalize: per-subgroup divergences + total ----------------
__device__ float blockReduceSum(float v, float* part) {
  const int lane = threadIdx.x & 31, wave = threadIdx.x >> 5;
  for (int off = 16; off >= 1; off >>= 1) v += __shfl_xor(v, off, 32);
  __syncthreads();
  if (lane == 0) part[wave] = v;
  __syncthreads();
  float r = 0.0f;
  for (int i = 0; i < 8; ++i) r += part[i];
  return r;
}

__global__ __launch_bounds__(256) void finalize_kernel(const float* __restrict__ f,
                                                       const float* __restrict__ gpot,
                                                       const float* __restrict__ fxx,
                                                       const float* __restrict__ fyy,
                                                       const int* __restrict__ sub,
                                                       const int* __restrict__ counts,
                                                       float* __restrict__ out) {
  __shared__ float part[8];
  const int tid = threadIdx.x;
  float s = 0.0f;
  for (int j = tid; j < B_N; j += 256) s += fyy[j];
  const float fyymean = blockReduceSum(s, part) * (1.0f / (float)B_N);

  float total = 0.0f;
  int nvalid = 0;
  for (int g = 0; g < G_N; ++g) {
    const int cg = counts[g];
    const float invc = 1.0f / (float)(cg > 1 ? cg : 1);
    float acc = 0.0f;
    for (int j = tid; j < B_N; j += 256) {
      const float a = (sub[j] == g) ? invc : 0.0f;
      acc += a * (f[g * B_N + j] - fxx[g * B_N + j])
           + gpot[g * B_N + j] * (1.0f / (float)B_N);
    }
    const float Ag = blockReduceSum(acc, part);
    const float S = Ag - fyymean;
    const bool valid = cg >= 2;
    const float sgl = valid ? S : 0.0f;
    if (tid == 0) out[1 + g] = sgl;
    total += sgl;
    nvalid += valid ? 1 : 0;
  }
  if (tid == 0) out[0] = total / (float)(nvalid > 1 ? nvalid : 1);
}

extern "C" void kernel_launch(void* const* d_in, const int* in_sizes, int n_in,
                              void* d_out, int out_size, void* d_ws, size_t ws_size,
                              hipStream_t stream) {
  const float* X  = (const float*)d_in[0];
  const int* sub  = (const int*)d_in[1];
  float* out = (float*)d_out;

  float* ws   = (float*)d_ws;
  float* C    = ws;                       // 2048*2048
  float* sqn  = C + (size_t)B_N * B_N;    // 2048
  float* f    = sqn + B_N;                // 8*2048
  float* g    = f + G_N * B_N;            // 8*2048   } g, fxx0, fyy0 contiguous -> one zero pass
  float* fxx0 = g + G_N * B_N;            // 8*2048   }
  float* fyy0 = fxx0 + G_N * B_N;         // 2048     }
  float* fxx1 = fyy0 + B_N;               // 8*2048
  float* fyy1 = fxx1 + G_N * B_N;         // 2048
  float* w    = fyy1 + B_N;               // 17*2048
  float* eps  = w + 17 * B_N;             // 64
  float* logcnt = eps + 64;               // 8
  int* counts   = (int*)(logcnt + G_N);   // 8
  unsigned* dmax = (unsigned*)(counts + G_N);

  sqnorm_kernel<<<B_N, 256, 0, stream>>>(X, sqn);
  counts_kernel<<<1, 256, 0, stream>>>(sub, counts, logcnt, dmax);
  gemm_cost_kernel<<<dim3(B_N / 16, B_N / 16), 32, 0, stream>>>(X, sqn, C, dmax);
  eps_kernel<<<1, 64, 0, stream>>>(dmax, eps);
  const int ZN = (2 * G_N + 1) * B_N;     // g + fxx0 + fyy0
  zero_kernel<<<(ZN + 255) / 256, 256, 0, stream>>>(g, ZN);

  for (int t = 0; t < 64; ++t) {
    float* fxx_cur  = (t & 1) ? fxx1 : fxx0;
    float* fxx_next = (t & 1) ? fxx0 : fxx1;
    float* fyy_cur  = (t & 1) ? fyy1 : fyy0;
    float* fyy_next = (t & 1) ? fyy0 : fyy1;
    // phase A: f from g; fxx, fyy self-averaged (17 slots)
    prepw_kernel<<<(17 * B_N) / 256, 256, 0, stream>>>(eps, t, 0, g, fxx_cur, fyy_cur,
                                                       sub, logcnt, w);
    lse_kernel<<<B_N / 4, 256, 0, stream>>>(C, w, eps, t, 17, 0, f,
                                            fxx_cur, fxx_next, fyy_cur, fyy_next);
    // phase B: g from fresh f (8 slots)
    prepw_kernel<<<(8 * B_N) / 256, 256, 0, stream>>>(eps, t, 1, f, fxx_cur, fyy_cur,
                                                      sub, logcnt, w);
    lse_kernel<<<B_N / 4, 256, 0, stream>>>(C, w, eps, t, 8, 1, g,
                                            nullptr, nullptr, nullptr, nullptr);
  }
  // after 64 iterations the final fxx/fyy live in buffer 0
  finalize_kernel<<<1, 256, 0, stream>>>(f, g, fxx0, fyy0, sub, counts, out);
}